// EdgeConv_58669253263648
// MI455X (gfx1250) — compile-verified
//
#include <hip/hip_runtime.h>

// EdgeConv (DGCNN block) for MI455X / gfx1250, wave32 + WMMA f32.
//
// Decomposition:
//   inner = X^T X via V_WMMA_F32_16X16X4_F32, 16-row stripes, dist stripe kept
//   entirely in 256KB LDS (CDNA5: 320KB/WGP) -> fused top-20 selection.
//   y[o,n,k] = ynb[o, idx[n,k]] + ypt[o,n]   with ynb = W1 @ x, ypt = (W2-W1) @ x
//   BN(batch stats) + LeakyReLU + max_k via per-point (max,min) + per-channel
//   deterministic two-stage sum/sumsq reduction.

typedef __attribute__((ext_vector_type(2))) float v2f;
typedef __attribute__((ext_vector_type(8))) float v8f;

#define BB   8
#define CC   64
#define NN   4096
#define KNN  20
#define OUTC 64

// ---------------------------------------------------------------- utilities
__device__ __forceinline__ unsigned long long shflx_u64(unsigned long long v, int m) {
  int lo = __shfl_xor((int)(v & 0xFFFFFFFFull), m, 32);
  int hi = __shfl_xor((int)(v >> 32), m, 32);
  return ((unsigned long long)(unsigned)hi << 32) | (unsigned)lo;
}
__device__ __forceinline__ unsigned long long wave_min_u64(unsigned long long v) {
#pragma unroll
  for (int m = 16; m; m >>= 1) {
    unsigned long long o = shflx_u64(v, m);
    v = (o < v) ? o : v;
  }
  return v;
}

// ------------------------------------------------- 1) squared norms per point
__global__ void sqnorm_kernel(const float* __restrict__ x, float* __restrict__ sq) {
  int i = blockIdx.x * blockDim.x + threadIdx.x;  // [0, B*N)
  int b = i >> 12;
  int n = i & (NN - 1);
  const float* xb = x + (size_t)b * CC * NN + n;
  float s = 0.f;
#pragma unroll 8
  for (int c = 0; c < CC; ++c) {
    float v = xb[(size_t)c * NN];
    s += v * v;
  }
  sq[i] = s;
}

// --------------------------- 2) fused distance-stripe (WMMA f32) + top-20 kNN
// One workgroup = 16 query rows vs all 4096 columns. 8 waves, each wave owns a
// 16-col slice per 128-col chunk. Distance keys land in LDS (16*4096*4=256KB).
__global__ void knn_kernel(const float* __restrict__ x, const float* __restrict__ sq,
                           int* __restrict__ idx_out) {
  extern __shared__ unsigned ldsD[];  // [16][4096] order-keys
  const int tid = threadIdx.x;
  const int lane = tid & 31, wave = tid >> 5;
  const int halfsel = lane >> 4;  // A/B frag layout: lanes 0-15 K={0,1}, 16-31 K={2,3}
  const int lm = lane & 15;
  const int b = blockIdx.x >> 8;
  const int n0 = (blockIdx.x & 255) << 4;

  const float* xb = x + (size_t)b * CC * NN;

  // Preload A fragments (the 16 query rows, all 64 K values) into registers.
  v2f a[16];
#pragma unroll
  for (int kk = 0; kk < 16; ++kk) {
    int c = kk * 4 + halfsel * 2;
    v2f t;
    t[0] = xb[(size_t)c * NN + (n0 + lm)];
    t[1] = xb[(size_t)(c + 1) * NN + (n0 + lm)];
    a[kk] = t;
  }

  // Sweep all columns: 32 chunks of 128 cols (8 waves x 16 cols each).
  for (int ch = 0; ch < 32; ++ch) {
    const int col = ch * 128 + wave * 16 + lm;
    v8f acc = {};
#pragma unroll
    for (int kk = 0; kk < 16; ++kk) {
      int c = kk * 4 + halfsel * 2;
      v2f bfrag;
      bfrag[0] = xb[(size_t)c * NN + col];
      bfrag[1] = xb[(size_t)(c + 1) * NN + col];
      acc = __builtin_amdgcn_wmma_f32_16x16x4_f32(false, a[kk], false, bfrag,
                                                  (short)0, acc, false, false);
    }
    // dist(n,m) = sq[n] + sq[m] - 2*inner; sq[n] is constant per row -> drop it
    // (order-preserving). Convert to monotone u32 key for exact selection.
    const float sv = sq[(size_t)b * NN + col];
#pragma unroll
    for (int r = 0; r < 8; ++r) {
      float d = sv - 2.0f * acc[r];
      unsigned u = __float_as_uint(d);
      unsigned key = u ^ ((u >> 31) ? 0xFFFFFFFFu : 0x80000000u);
      ldsD[(size_t)(r + halfsel * 8) * NN + col] = key;  // D: VGPR r = rows r / r+8
    }
  }
  __syncthreads();

  // Top-20 per row: wave w handles rows w and w+8. Each lane keeps a sorted
  // top-20 of its 128 strided candidates, then 20 rounds of wave-min merge.
  for (int rr = 0; rr < 2; ++rr) {
    const int row = wave + rr * 8;
    const unsigned* drow = ldsD + (size_t)row * NN;
    unsigned long long loc[KNN];
#pragma unroll
    for (int i = 0; i < KNN; ++i) loc[i] = ~0ull;
    for (int c = lane; c < NN; c += 32) {
      unsigned long long key = ((unsigned long long)drow[c] << 32) | (unsigned)c;
      if (key < loc[KNN - 1]) {
        int p = KNN - 1;
        while (p > 0 && loc[p - 1] > key) { loc[p] = loc[p - 1]; --p; }
        loc[p] = key;
      }
    }
    int ptr = 0;
    int* orow = idx_out + ((size_t)b * NN + (n0 + row)) * KNN;
    for (int j = 0; j < KNN; ++j) {
      unsigned long long cand = (ptr < KNN) ? loc[ptr] : ~0ull;
      unsigned long long g = wave_min_u64(cand);
      unsigned long long msk = __ballot(cand == g);
      int winner = __ffsll((long long)msk) - 1;
      if (lane == winner) ++ptr;
      if (lane == 0) orow[j] = (int)(g & 0xFFFFFFFFull);
    }
  }
}

// ---------------- 3) per-point transforms ynb = W1@x, ypt = (W2-W1)@x (WMMA)
__global__ void transform_kernel(const float* __restrict__ x, const float* __restrict__ W,
                                 float* __restrict__ ynb, float* __restrict__ ypt) {
  const int tid = threadIdx.x;
  const int lane = tid & 31;
  const int halfsel = lane >> 4, lm = lane & 15;
  const int w = blockIdx.x * 8 + (tid >> 5);  // 8192 waves total
  const int b = w >> 10;
  const int o0 = ((w >> 8) & 3) * 16;
  const int m0 = (w & 255) * 16;

  const float* xb = x + (size_t)b * CC * NN;
  const int oA = o0 + lm;

  v2f a1[16], ad[16];
#pragma unroll
  for (int kk = 0; kk < 16; ++kk) {
    int c = kk * 4 + halfsel * 2;
    float w1x = W[(size_t)oA * 128 + c];
    float w1y = W[(size_t)oA * 128 + c + 1];
    v2f t1; t1[0] = w1x; t1[1] = w1y;
    v2f t2;
    t2[0] = W[(size_t)oA * 128 + 64 + c] - w1x;
    t2[1] = W[(size_t)oA * 128 + 64 + c + 1] - w1y;
    a1[kk] = t1; ad[kk] = t2;
  }

  v8f acc1 = {}, acc2 = {};
#pragma unroll
  for (int kk = 0; kk < 16; ++kk) {
    int c = kk * 4 + halfsel * 2;
    v2f bfrag;
    bfrag[0] = xb[(size_t)c * NN + (m0 + lm)];
    bfrag[1] = xb[(size_t)(c + 1) * NN + (m0 + lm)];
    acc1 = __builtin_amdgcn_wmma_f32_16x16x4_f32(false, a1[kk], false, bfrag,
                                                 (short)0, acc1, false, false);
    acc2 = __builtin_amdgcn_wmma_f32_16x16x4_f32(false, ad[kk], false, bfrag,
                                                 (short)0, acc2, false, false);
  }
#pragma unroll
  for (int r = 0; r < 8; ++r) {
    int oo = o0 + r + halfsel * 8;
    size_t off = ((size_t)b * OUTC + oo) * NN + (m0 + lm);
    ynb[off] = acc1[r];
    ypt[off] = acc2[r];
  }
}

// ------- 4) gather neighbors, per-point max/min over k, partial BN statistics
__global__ void gather_stats_kernel(const float* __restrict__ ynb, const float* __restrict__ ypt,
                                    const int* __restrict__ idx, float* __restrict__ mx,
                                    float* __restrict__ mn, float* __restrict__ part) {
  __shared__ float rs[256], rss[256];
  const int bi = blockIdx.x;          // B*64*16 blocks
  const int b = bi >> 10;
  const int o = (bi >> 4) & 63;
  const int ck = bi & 15;
  const int n = ck * 256 + threadIdx.x;

  const float* row = ynb + ((size_t)b * OUTC + o) * NN;  // 16KB, L0/L2 resident
  const float p = ypt[((size_t)b * OUTC + o) * NN + n];
  const int* ir = idx + ((size_t)b * NN + n) * KNN;

  float s = 0.f, ss = 0.f;
  float amx = -3.402823466e38f, amn = 3.402823466e38f;
#pragma unroll
  for (int k = 0; k < KNN; ++k) {
    float y = row[ir[k]] + p;
    s += y; ss += y * y;
    amx = fmaxf(amx, y); amn = fminf(amn, y);
  }
  size_t off = ((size_t)b * OUTC + o) * NN + n;
  mx[off] = amx; mn[off] = amn;

  rs[threadIdx.x] = s; rss[threadIdx.x] = ss;
  __syncthreads();
  for (int st = 128; st > 0; st >>= 1) {
    if (threadIdx.x < st) {
      rs[threadIdx.x] += rs[threadIdx.x + st];
      rss[threadIdx.x] += rss[threadIdx.x + st];
    }
    __syncthreads();
  }
  if (threadIdx.x == 0) {
    size_t pi = ((size_t)o * 128 + b * 16 + ck) * 2;
    part[pi] = rs[0]; part[pi + 1] = rss[0];
  }
}

// ---------------------------- 5) finalize BN: per-channel scale/shift factors
__global__ void finalize_kernel(const float* __restrict__ part, const float* __restrict__ gamma,
                                const float* __restrict__ beta, float* __restrict__ scsh) {
  __shared__ float rs[128], rss[128];
  const int o = blockIdx.x, t = threadIdx.x;
  rs[t] = part[((size_t)o * 128 + t) * 2];
  rss[t] = part[((size_t)o * 128 + t) * 2 + 1];
  __syncthreads();
  for (int st = 64; st > 0; st >>= 1) {
    if (t < st) { rs[t] += rs[t + st]; rss[t] += rss[t + st]; }
    __syncthreads();
  }
  if (t == 0) {
    const float cnt = (float)((size_t)BB * NN * KNN);
    float mean = rs[0] / cnt;
    float var = rss[0] / cnt - mean * mean;  // biased, as jnp.var
    float inv = rsqrtf(var + 1e-5f);
    float sc = gamma[o] * inv;
    float sh = beta[o] - mean * sc;
    scsh[2 * o] = sc; scsh[2 * o + 1] = sh;
  }
}

// ------------------------- 6) BN affine + LeakyReLU on the extremal neighbor
__global__ void output_kernel(const float* __restrict__ mx, const float* __restrict__ mn,
                              const float* __restrict__ scsh, float* __restrict__ out) {
  int i = blockIdx.x * blockDim.x + threadIdx.x;  // [0, B*64*N)
  int o = (i >> 12) & 63;
  float sc = scsh[2 * o], sh = scsh[2 * o + 1];
  float v = (sc >= 0.f) ? mx[i] : mn[i];  // monotone affine: max_k commutes
  float y = sc * v + sh;
  out[i] = (y >= 0.f) ? y : 0.2f * y;
}

// --------------------------------------------------------------------- host
extern "C" void kernel_launch(void* const* d_in, const int* in_sizes, int n_in,
                              void* d_out, int out_size, void* d_ws, size_t ws_size,
                              hipStream_t stream) {
  const float* x = (const float*)d_in[0];      // [8,64,4096]
  const float* W = (const float*)d_in[1];      // [64,128]
  const float* gamma = (const float*)d_in[2];  // [64]
  const float* beta = (const float*)d_in[3];   // [64]
  float* out = (float*)d_out;                  // [8,64,4096]

  char* p = (char*)d_ws;
  float* sq = (float*)p;   p += (size_t)BB * NN * 4;
  int* idx = (int*)p;      p += (size_t)BB * NN * KNN * 4;
  float* ynb = (float*)p;  p += (size_t)BB * OUTC * NN * 4;
  float* ypt = (float*)p;  p += (size_t)BB * OUTC * NN * 4;
  float* mxb = (float*)p;  p += (size_t)BB * OUTC * NN * 4;
  float* mnb = (float*)p;  p += (size_t)BB * OUTC * NN * 4;
  float* part = (float*)p; p += (size_t)OUTC * 128 * 2 * 4;
  float* scsh = (float*)p; p += (size_t)OUTC * 2 * 4;
  (void)in_sizes; (void)n_in; (void)out_size; (void)ws_size;

  sqnorm_kernel<<<(BB * NN) / 256, 256, 0, stream>>>(x, sq);
  // 256KB dynamic LDS: the full 16x4096 distance stripe lives on-chip.
  knn_kernel<<<BB * (NN / 16), 256, 16 * NN * sizeof(unsigned), stream>>>(x, sq, idx);
  transform_kernel<<<(BB * 4 * 256) / 8, 256, 0, stream>>>(x, W, ynb, ypt);
  gather_stats_kernel<<<BB * OUTC * 16, 256, 0, stream>>>(ynb, ypt, idx, mxb, mnb, part);
  finalize_kernel<<<OUTC, 128, 0, stream>>>(part, gamma, beta, scsh);
  output_kernel<<<(BB * OUTC * NN) / 256, 256, 0, stream>>>(mxb, mnb, scsh, out);
}